// LinODECell_1288490189421
// MI455X (gfx1250) — compile-verified
//
#include <hip/hip_runtime.h>

typedef float v2f __attribute__((ext_vector_type(2)));
typedef float v4f __attribute__((ext_vector_type(4)));
typedef float v8f __attribute__((ext_vector_type(8)));

#define K_TERMS 24
#define WAVES_PER_BLOCK 4
#define LDS_STRIDE 68   // 64 + 4 floats pad: lane n -> bank (4n + m) % 64, conflict-free

// 1/k for the Taylor coefficient recurrence (index 0 unused)
__device__ __constant__ float g_invk[K_TERMS + 1] = {
    0.0f,
    1.0f/1.0f,  1.0f/2.0f,  1.0f/3.0f,  1.0f/4.0f,  1.0f/5.0f,  1.0f/6.0f,
    1.0f/7.0f,  1.0f/8.0f,  1.0f/9.0f,  1.0f/10.0f, 1.0f/11.0f, 1.0f/12.0f,
    1.0f/13.0f, 1.0f/14.0f, 1.0f/15.0f, 1.0f/16.0f, 1.0f/17.0f, 1.0f/18.0f,
    1.0f/19.0f, 1.0f/20.0f, 1.0f/21.0f, 1.0f/22.0f, 1.0f/23.0f, 1.0f/24.0f
};

// One wave32 handles 16 batch elements: y[64x16] = expm(dt*A)[64x64] @ x0[64x16]
// via Taylor series, each step V <- A@V done with 64x V_WMMA_F32_16X16X4_F32.
__global__ __launch_bounds__(32 * WAVES_PER_BLOCK)
void linode_expm_wmma(const float* __restrict__ dt,
                      const float* __restrict__ x0,
                      const float* __restrict__ A,
                      float* __restrict__ out,
                      int num_groups)
{
    const int lane = threadIdx.x & 31;
    const int wave = threadIdx.x >> 5;
    const int group = blockIdx.x * WAVES_PER_BLOCK + wave;
    if (group >= num_groups) return;   // never taken (grid divides exactly); wave-uniform

    const int n = lane & 15;   // batch within the 16-wide group (WMMA N index)
    const int h = lane >> 4;   // lane half (WMMA K/M striping)
    const long b = (long)group * 16 + n;   // this lane's batch row

    // Wave-private LDS staging panel, layout [batch n][state m], padded stride.
    __shared__ float lds[WAVES_PER_BLOCK][16][LDS_STRIDE];
    float (*V)[LDS_STRIDE] = lds[wave];

    // ---- Load A (shared 64x64) into registers in WMMA A-operand layout ----
    // Tile (r, kb) covers rows 16r..16r+15, cols 4kb..4kb+3.
    // A-layout (16x4 f32): lane l holds M = l&15; VGPR j holds K = 2*(l>>4) + j.
    v2f a[64];
    #pragma unroll
    for (int r = 0; r < 4; ++r) {
        #pragma unroll
        for (int kb = 0; kb < 16; ++kb) {
            const float* p = A + (size_t)(16*r + n) * 64 + 4*kb + 2*h;
            a[r*16 + kb] = *(const v2f*)p;   // global_load_b64
        }
    }

    const float dtv = dt[b];

    // ---- Load x0 panel in C/D layout: tile t, VGPR g -> row m = 16t + 8h + g ----
    v8f y[4], v[4];
    #pragma unroll
    for (int t = 0; t < 4; ++t) {
        const float* p = x0 + b * 64 + 16*t + 8*h;
        v[t] = *(const v8f*)p;               // 2x global_load_b128
        y[t] = v[t];                         // k=0 term: y = x0
    }

    float coef = 1.0f;

    #pragma unroll 1   // keep rolled: inner body already ~64 wmma + ds ops
    for (int k = 1; k <= K_TERMS; ++k) {
        // ---- v (C/D layout) -> LDS panel, row per batch n, column per state m: two b128 stores per tile ----
        #pragma unroll
        for (int t = 0; t < 4; ++t) {
            v4f lo, hi;
            #pragma unroll
            for (int e = 0; e < 4; ++e) { lo[e] = v[t][e]; hi[e] = v[t][e + 4]; }
            *(v4f*)&V[n][16*t + 8*h]     = lo;
            *(v4f*)&V[n][16*t + 8*h + 4] = hi;
        }
        // Same-wave DS pipe is in-order; wait + compiler barrier before re-reading.
        asm volatile("s_wait_dscnt 0" ::: "memory");

        #pragma unroll
        for (int t = 0; t < 4; ++t) v[t] = (v8f){0,0,0,0,0,0,0,0};

        // ---- V <- A @ V : stream B k-blocks from LDS, 4 wmma each ----
        // B-layout (4x16 f32): lane l holds N = l&15; VGPR j holds K = 2*(l>>4)+j,
        // i.e. global row 4*kb + 2h + j -> contiguous pair in LDS -> ds_load_b64.
        #pragma unroll
        for (int kb = 0; kb < 16; ++kb) {
            v2f bb = *(const v2f*)&V[n][4*kb + 2*h];
            #pragma unroll
            for (int r = 0; r < 4; ++r) {
                v[r] = __builtin_amdgcn_wmma_f32_16x16x4_f32(
                    /*neg_a=*/false, a[r*16 + kb],
                    /*neg_b=*/false, bb,
                    /*c_mod=*/(short)0, v[r],
                    /*reuse_a=*/false, /*reuse_b=*/false);
            }
        }

        // ---- y += (dt^k / k!) * v  (per-lane coefficient, batch = n) ----
        coef *= dtv * g_invk[k];
        #pragma unroll
        for (int t = 0; t < 4; ++t)
            #pragma unroll
            for (int e = 0; e < 8; ++e)
                y[t][e] += coef * v[t][e];
    }

    // ---- Store result: C/D layout maps back to row-major out[b][m] ----
    #pragma unroll
    for (int t = 0; t < 4; ++t) {
        float* p = out + b * 64 + 16*t + 8*h;
        v4f lo, hi;
        #pragma unroll
        for (int e = 0; e < 4; ++e) { lo[e] = y[t][e]; hi[e] = y[t][e + 4]; }
        *(v4f*)p       = lo;                 // global_store_b128
        *(v4f*)(p + 4) = hi;
    }
}

extern "C" void kernel_launch(void* const* d_in, const int* in_sizes, int n_in,
                              void* d_out, int out_size, void* d_ws, size_t ws_size,
                              hipStream_t stream) {
    const float* dt = (const float*)d_in[0];   // [B]
    const float* x0 = (const float*)d_in[1];   // [B, 64]
    const float* A  = (const float*)d_in[2];   // [64, 64]
    float* out = (float*)d_out;                // [B, 64]

    const int Btot   = in_sizes[0];            // 16384
    const int groups = Btot / 16;              // 1024 waves of work

    dim3 block(32 * WAVES_PER_BLOCK);          // 128 threads = 4 waves
    dim3 grid((groups + WAVES_PER_BLOCK - 1) / WAVES_PER_BLOCK);
    linode_expm_wmma<<<grid, block, 0, stream>>>(dt, x0, A, out, groups);
}